// PlasticLeakyRNNCell_46926812676405
// MI455X (gfx1250) — compile-verified
//
#include <hip/hip_runtime.h>

#define B_  64
#define I_  128
#define A_  16
#define H_  1024
#define E_  819          // int(0.8 * 1024)
#define DTC 0.02f
#define AX  0.2f         // dt / tau_x
#define AW  0.02f        // dt / tau_w

typedef float v2f __attribute__((ext_vector_type(2)));
typedef float v4f __attribute__((ext_vector_type(4)));
typedef float v8f __attribute__((ext_vector_type(8)));

// ---------------------------------------------------------------------------
// K1: ffwd[b,h] = b_h[h] + relu(x[b])·|W_x[h]| + relu(aux[b])·|W_aux[h]|
// ---------------------------------------------------------------------------
__global__ __launch_bounds__(256) void ffwd_kernel(
    const float* __restrict__ x, const float* __restrict__ aux,
    const float* __restrict__ Wx, const float* __restrict__ Waux,
    const float* __restrict__ bh, float* __restrict__ ffwd) {
  int idx = blockIdx.x * 256 + threadIdx.x;      // [0, B*H)
  int b = idx >> 10, h = idx & (H_ - 1);
  const v4f* xr = (const v4f*)(x + b * I_);
  const v4f* wr = (const v4f*)(Wx + h * I_);
  float acc = bh[h];
#pragma unroll
  for (int i = 0; i < I_ / 4; ++i) {
    v4f xv = xr[i], wv = wr[i];
    acc += fmaxf(xv.x, 0.f) * fabsf(wv.x) + fmaxf(xv.y, 0.f) * fabsf(wv.y)
         + fmaxf(xv.z, 0.f) * fabsf(wv.z) + fmaxf(xv.w, 0.f) * fabsf(wv.w);
  }
  const v4f* ar = (const v4f*)(aux + b * A_);
  const v4f* wa = (const v4f*)(Waux + h * A_);
#pragma unroll
  for (int i = 0; i < A_ / 4; ++i) {
    v4f av = ar[i], wv = wa[i];
    acc += fmaxf(av.x, 0.f) * fabsf(wv.x) + fmaxf(av.y, 0.f) * fabsf(wv.y)
         + fmaxf(av.z, 0.f) * fabsf(wv.z) + fmaxf(av.w, 0.f) * fabsf(wv.w);
  }
  ffwd[idx] = acc;
}

// ---------------------------------------------------------------------------
// K2: 256-thread block = 8 wave32s, all on batch b, covering 128 rows of i.
// Signed output vector v[j] = sign(j)*output[b,j] staged once per block in LDS.
// Each wave: 16-row tile, total_i = sum_j wh[b,i,j]*v[j] (minus diagonal) via
// V_WMMA_F32_16X16X4_F32 (fp32-exact; B operand broadcast across N columns).
// Epilogue: new_state = state*(1-AX) + (total+ffwd)*AX ; new_output = retanh.
// ---------------------------------------------------------------------------
__global__ __launch_bounds__(256) void recur_kernel(
    const float* __restrict__ wh, const float* __restrict__ output,
    const float* __restrict__ state, const float* __restrict__ ffwd,
    float* __restrict__ out_state, float* __restrict__ out_output) {
  __shared__ __align__(16) float vsh[H_];
  __shared__ float tot[8][16];

  const int tid  = threadIdx.x;
  const int lane = tid & 31;                     // EXEC all-1s inside the wave
  const int wid  = tid >> 5;                     // 0..7
  const int b    = blockIdx.x >> 3;              // 8 blocks per batch
  const int i0   = ((blockIdx.x & 7) << 7) + (wid << 4);

  // Stage signed output vector once per block.
  const float* orow = output + (size_t)b * H_;
  for (int j = tid; j < H_; j += 256) {
    float o = orow[j];
    vsh[j] = (j < E_) ? o : -o;
  }
  __syncthreads();

  // A-matrix per-lane layout (32-bit 16x4): lanes 0-15 -> K=k,k+1 ; 16-31 -> k+2,k+3
  const int mrow  = lane & 15;
  const int khalf = (lane >> 4) << 1;            // 0 or 2
  const float* wrow = wh + ((size_t)b * H_ + (i0 + mrow)) * (size_t)H_;

  v8f c = {};
  for (int k0 = 0; k0 < H_; k0 += 4) {
    const int col = k0 + khalf;
    v2f a  = *(const v2f*)(wrow + col);          // wh tile (RT: prime L2 for K3)
    v2f bb = *(const v2f*)(vsh + col);           // broadcast across N columns
    c = __builtin_amdgcn_wmma_f32_16x16x4_f32(
        /*neg_a=*/false, a, /*neg_b=*/false, bb,
        /*c_mod=*/(short)0, c, /*reuse_a=*/false, /*reuse_b=*/false);
  }

  // Diagonal removal (mask zeroes i==j): lane L holds row i0+L's correction.
  const int dcol = i0 + mrow;
  const float dia = wrow[dcol] * vsh[dcol];

  // D layout: VGPR r, lanes 0-15 -> M=r ; lanes 16-31 -> M=8+r (all N identical).
  if (lane == 0) {
#pragma unroll
    for (int r = 0; r < 8; ++r) tot[wid][r] = c[r];
  }
  if (lane == 16) {
#pragma unroll
    for (int r = 0; r < 8; ++r) tot[wid][8 + r] = c[r];
  }
  __syncthreads();

  if (lane < 16) {
    const int h = i0 + lane;
    const size_t o = (size_t)b * H_ + h;
    const float t  = tot[wid][lane] - dia;
    const float ns = state[o] * (1.f - AX) + (t + ffwd[o]) * AX;
    out_state[o]  = ns;
    out_output[o] = tanhf(fmaxf(ns, 0.f));       // retanh
  }
}

// ---------------------------------------------------------------------------
// K3: new_w = clip(|W_h|*AW + wh*(1-AW) + DT*da_b*|kappa|*no_i*no_j, 0, 1)
// float4 over [B,H,H]. wh read NT (last use, may hit K2's L2 leftovers);
// new_w stored NT (never re-read — keep it out of L2).
// ---------------------------------------------------------------------------
__global__ __launch_bounds__(256) void plast_kernel(
    const float* __restrict__ wh, const float* __restrict__ Wh,
    const float* __restrict__ kap, const float* __restrict__ da,
    const float* __restrict__ nout, float* __restrict__ out_w) {
  const size_t idx = (size_t)blockIdx.x * 256 + threadIdx.x;  // [0, 2^24)
  const int b   = (int)(idx >> 18);              // H*H/4 = 2^18 float4 per batch
  const int rem = (int)(idx & 0x3FFFF);
  const int i   = rem >> 8;                      // H/4 = 256 float4 per row
  const int c4  = rem & 255;

  const v4f whv = __builtin_nontemporal_load(&((const v4f*)wh)[idx]);
  const v4f Whv = ((const v4f*)Wh)[(i << 8) + c4];
  const v4f kv  = ((const v4f*)kap)[(i << 8) + c4];
  const v4f nj  = ((const v4f*)nout)[(b << 8) + c4];
  const float ni = nout[(b << 10) + i];
  const float cf = DTC * da[b] * ni;

  v4f r;
  r.x = fminf(fmaxf(fabsf(Whv.x) * AW + whv.x * (1.f - AW) + cf * fabsf(kv.x) * nj.x, 0.f), 1.f);
  r.y = fminf(fmaxf(fabsf(Whv.y) * AW + whv.y * (1.f - AW) + cf * fabsf(kv.y) * nj.y, 0.f), 1.f);
  r.z = fminf(fmaxf(fabsf(Whv.z) * AW + whv.z * (1.f - AW) + cf * fabsf(kv.z) * nj.z, 0.f), 1.f);
  r.w = fminf(fmaxf(fabsf(Whv.w) * AW + whv.w * (1.f - AW) + cf * fabsf(kv.w) * nj.w, 0.f), 1.f);
  __builtin_nontemporal_store(r, &((v4f*)out_w)[idx]);
}

// ---------------------------------------------------------------------------
extern "C" void kernel_launch(void* const* d_in, const int* in_sizes, int n_in,
                              void* d_out, int out_size, void* d_ws, size_t ws_size,
                              hipStream_t stream) {
  (void)in_sizes; (void)n_in; (void)out_size; (void)ws_size;
  const float* x      = (const float*)d_in[0];
  const float* state  = (const float*)d_in[1];
  const float* output = (const float*)d_in[2];
  const float* wh     = (const float*)d_in[3];
  const float* da     = (const float*)d_in[4];
  const float* aux    = (const float*)d_in[5];
  const float* Wx     = (const float*)d_in[6];
  const float* Waux   = (const float*)d_in[7];
  const float* Wh     = (const float*)d_in[8];
  const float* bh     = (const float*)d_in[9];
  const float* kap    = (const float*)d_in[10];
  // d_in[11] = mask_h: folded analytically (sign by column, zero diagonal)

  float* out        = (float*)d_out;
  float* out_state  = out;                       // [B,H]
  float* out_output = out + B_ * H_;             // [B,H]
  float* out_w      = out + 2 * B_ * H_;         // [B,H,H]
  float* ffwd       = (float*)d_ws;              // [B,H] scratch

  ffwd_kernel<<<(B_ * H_) / 256, 256, 0, stream>>>(x, aux, Wx, Waux, bh, ffwd);
  recur_kernel<<<B_ * 8, 256, 0, stream>>>(wh, output, state, ffwd,
                                           out_state, out_output);
  plast_kernel<<<(B_ * H_ * H_ / 4) / 256, 256, 0, stream>>>(wh, Wh, kap, da,
                                                             out_output, out_w);
}